// HaarDown_19258633355741
// MI455X (gfx1250) — compile-verified
//
#include <hip/hip_runtime.h>
#include <hip/hip_bf16.h>

// Depthwise 2x2 Haar high-pass:
//   out[n,i,j] = 0.5*((x[i,j]-x[i,j+1]) - (x[i+1,j]-x[i+1,j+1]))
// Memory-bound (AI ~0.9 FLOP/B, HBM floor ~44us at 23.3 TB/s).
// Strategy: async-stage row tiles into LDS (exactly-once HBM reads),
// register-rolling vertical reuse, non-temporal stores.

#ifndef __has_builtin
#define __has_builtin(x) 0
#endif

#if __has_builtin(__builtin_amdgcn_global_load_async_to_lds_b128) && \
    __has_builtin(__builtin_amdgcn_s_wait_asynccnt)
#define HAAR_ASYNC 1
#else
#define HAAR_ASYNC 0
#endif

#define HW_W   512
#define HW_H   512
#define OW     511
#define OH     511
#define TR     16            // output rows per strip
#define STRIPS 32            // ceil(511/16)
#define NTHREADS 256         // 8 waves (wave32)

// Must be GCC-style vector_size (matches the builtin's prototype exactly).
typedef int v4i __attribute__((vector_size(16)));

__global__ __launch_bounds__(NTHREADS)
void haar_highpass_kernel(const float* __restrict__ x, float* __restrict__ out) {
    // (TR+1) input rows of 512 floats, +4 pad so the j+4 edge read stays in bounds
    __shared__ float smem[(TR + 1) * HW_W + 4];

    const int tid  = threadIdx.x;
    const int n    = blockIdx.x / STRIPS;          // image index (B*C = 512)
    const int s    = blockIdx.x % STRIPS;          // strip index
    const int row0 = s * TR;                       // first output row of strip
    const int orows = min(TR, OH - row0);          // 16, or 15 for last strip
    const int irows = orows + 1;                   // input rows to stage (<= H)

    const float* gimg = x   + (size_t)n * (HW_H * HW_W);
    float*       oimg = out + (size_t)n * (OH * OW);

    // ---- Stage irows x 512 floats into LDS with async 128-bit transfers ----
    const int nvec = irows * (HW_W / 4);           // float4 chunks
    for (int i = tid; i < nvec; i += NTHREADS) {
        const int r  = i >> 7;                     // HW_W/4 == 128
        const int c4 = (i & 127) * 4;
        const float* gp = gimg + (size_t)(row0 + r) * HW_W + c4;
        float*       lp = &smem[r * HW_W + c4];
#if HAAR_ASYNC
        __builtin_amdgcn_global_load_async_to_lds_b128(
            (__attribute__((address_space(1))) v4i*)gp,
            (__attribute__((address_space(3))) v4i*)lp,
            0, 0);
#else
        *(float4*)lp = *(const float4*)gp;
#endif
    }
#if HAAR_ASYNC
    __builtin_amdgcn_s_wait_asynccnt(0);
#endif
    __syncthreads();

    // ---- Compute: thread = (row-half, 4-column group); roll rows in registers ----
    const int half = tid >> 7;                     // 0 or 1
    const int jb   = (tid & 127) * 4;              // output column base, 0..508
    const int rb   = half * 8;
    const int re   = min(orows, rb + 8);

    if (rb < re) {
        // previous-row window: 5 values x[rb][jb..jb+4]
        float4 a4 = *(const float4*)&smem[rb * HW_W + jb];
        float  ae = smem[rb * HW_W + jb + 4];
        float a0 = a4.x, a1 = a4.y, a2 = a4.z, a3 = a4.w;

        for (int r = rb; r < re; ++r) {
            const float4 b4 = *(const float4*)&smem[(r + 1) * HW_W + jb];
            const float  be = smem[(r + 1) * HW_W + jb + 4];

            const float o0 = 0.5f * ((a0 - a1) - (b4.x - b4.y));
            const float o1 = 0.5f * ((a1 - a2) - (b4.y - b4.z));
            const float o2 = 0.5f * ((a2 - a3) - (b4.z - b4.w));
            const float o3 = 0.5f * ((a3 - ae) - (b4.w - be));

            float* op = oimg + (size_t)(row0 + r) * OW + jb;
            if (jb + 3 < OW) {                     // full 4-wide group
                __builtin_nontemporal_store(o0, op + 0);
                __builtin_nontemporal_store(o1, op + 1);
                __builtin_nontemporal_store(o2, op + 2);
                __builtin_nontemporal_store(o3, op + 3);
            } else {                               // tail group: cols 508..510
                if (jb + 0 < OW) __builtin_nontemporal_store(o0, op + 0);
                if (jb + 1 < OW) __builtin_nontemporal_store(o1, op + 1);
                if (jb + 2 < OW) __builtin_nontemporal_store(o2, op + 2);
            }

            a0 = b4.x; a1 = b4.y; a2 = b4.z; a3 = b4.w; ae = be;
        }
    }
}

extern "C" void kernel_launch(void* const* d_in, const int* in_sizes, int n_in,
                              void* d_out, int out_size, void* d_ws, size_t ws_size,
                              hipStream_t stream) {
    const float* x = (const float*)d_in[0];
    float* out = (float*)d_out;
    const int n_imgs = in_sizes[0] / (HW_H * HW_W);   // B*C = 512
    dim3 grid(n_imgs * STRIPS);
    dim3 block(NTHREADS);
    haar_highpass_kernel<<<grid, block, 0, stream>>>(x, out);
}